// RotorQuantMSE_38190849196138
// MI455X (gfx1250) — compile-verified
//
#include <hip/hip_runtime.h>

// RotorQuantMSE for MI455X (gfx1250, wave32).
// Memory-bound (~75 MB traffic, ~90 MFLOP @ 23.3 TB/s -> ~3.2 us): one fused streaming pass.
// Per-group 3x3 rotations run on V_WMMA_F32_16X16X4_F32:
//   forward:  A = M_g (3x3 zero-padded 16x4), B = x-tile group columns (4x16, col = x-row)
//   backward: A = M_g^T, B = quantized D (comp2 hops lo->hi half-wave via shfl_xor 16)
// A operands are pre-packed in WMMA lane order so the hot loop is branchless:
// 2x ds_load_b64 (A fwd/bwd) + 2x ds_load_b32 (B) + 2x wmma + quantize + b96 stores.

typedef __attribute__((ext_vector_type(2))) float v2f;
typedef __attribute__((ext_vector_type(8))) float v8f;

#define DD    768
#define GG    256
#define ROWS  16
#define LROW  772   // padded LDS row stride in floats (768 % 64 banks == 0 -> pad by 4)

// ---------- full geometric product (Cayley table, same construction as reference) ----------
__device__ __forceinline__ void gprod(const float a[8], const float b[8], float c[8]) {
  const int ORDER[8] = {0, 1, 2, 4, 3, 5, 6, 7};   // basis index -> blade bitmask
  const int IDXOF[8] = {0, 1, 2, 4, 3, 5, 6, 7};   // blade bitmask -> basis index
  for (int k = 0; k < 8; ++k) c[k] = 0.f;
  for (int i = 0; i < 8; ++i) {
    for (int j = 0; j < 8; ++j) {
      int ma = ORDER[i], mb = ORDER[j];
      int s = 0;
      for (int n = ma >> 1; n; n >>= 1) s += __popc(n & mb);
      float sgn = (s & 1) ? -1.f : 1.f;
      c[IDXOF[ma ^ mb]] += sgn * a[i] * b[j];
    }
  }
}

// Per group g: rotation matrix M[row][col] = (R e_col ~R)[row], emitted pre-packed in
// WMMA-A lane order. Layout (32 floats per group), e = 0..3 (row index, e==3 is the
// zero row that A rows 3..15 clamp onto):
//   [dir*16 + hi*8 + e*2 + {0,1}]
//   dir=0 (fwd, A=M):    lo: (M[e][0], M[e][1])   hi: (M[e][2], 0)
//   dir=1 (bwd, A=M^T):  lo: (M[0][e], M[1][e])   hi: (M[2][e], 0)
__global__ void rotprep_kernel(const float* __restrict__ rotors,
                               float* __restrict__ Mpack, int G) {
  int g = blockIdx.x * blockDim.x + threadIdx.x;
  if (g >= G) return;
  const float REV[8] = {1.f, 1.f, 1.f, 1.f, -1.f, -1.f, -1.f, -1.f};
  float R[8], Rrev[8];
  for (int k = 0; k < 8; ++k) { R[k] = rotors[g * 8 + k]; Rrev[k] = R[k] * REV[k]; }
  float M[3][3];
  for (int col = 0; col < 3; ++col) {
    float e[8] = {0.f, 0.f, 0.f, 0.f, 0.f, 0.f, 0.f, 0.f};
    e[1 + col] = 1.f;
    float t[8], w[8];
    gprod(R, e, t);
    gprod(t, Rrev, w);
    for (int row = 0; row < 3; ++row) M[row][col] = w[1 + row];
  }
  float* P = Mpack + (size_t)g * 32;
  for (int k = 0; k < 32; ++k) P[k] = 0.f;
  for (int e = 0; e < 3; ++e) {
    P[ 0 + e * 2 + 0] = M[e][0];  P[ 0 + e * 2 + 1] = M[e][1];   // fwd lo
    P[ 8 + e * 2 + 0] = M[e][2];                                  // fwd hi
    P[16 + e * 2 + 0] = M[0][e];  P[16 + e * 2 + 1] = M[1][e];   // bwd lo
    P[24 + e * 2 + 0] = M[2][e];                                  // bwd hi
  }
}

__device__ __forceinline__ void quantize1(float v, float c0, float cstep,
                                          float& q, float& fi) {
  // uniform centroids; argmin ties -> lower index => round-half-down
  float t = (v + 1.0f) * 7.5f;
  int i = (int)ceilf(t - 0.5f);
  i = i < 0 ? 0 : (i > 15 ? 15 : i);
  fi = (float)i;
  q  = fmaf(fi, cstep, c0);
}

__global__ __launch_bounds__(256) void rq_main_kernel(
    const float* __restrict__ x, const float* __restrict__ centroids,
    const float* __restrict__ Mpack,
    float* __restrict__ xhat, float* __restrict__ idxo, float* __restrict__ normo) {
  __shared__ float ldsx[ROWS * LROW];   // 48.25 KB x-tile (bank-conflict padded)
  __shared__ float ldsP[GG * 32];       // 32 KB pre-packed A operands
  __shared__ float part[ROWS * 16];
  __shared__ float snorm[ROWS];
  __shared__ float sinv[ROWS];

  const int tid = threadIdx.x;
  const long long rowBase = (long long)blockIdx.x * ROWS;
  const float c0    = centroids[0];
  const float cstep = centroids[1] - centroids[0];

  // ---- cooperative tile + packed-M load (coalesced float4) ----
  {
    const float4* __restrict__ x4 = (const float4*)(x + rowBase * DD);
    float4* l4 = (float4*)ldsx;                 // LROW/4 = 193 float4 per row
    for (int k = tid; k < ROWS * (DD / 4); k += 256) {
      int r = k / (DD / 4), c4 = k % (DD / 4);
      l4[r * (LROW / 4) + c4] = x4[k];
    }
    const float4* __restrict__ p4 = (const float4*)Mpack;
    float4* lp4 = (float4*)ldsP;
    for (int k = tid; k < GG * 8; k += 256) lp4[k] = p4[k];
  }
  __syncthreads();

  // ---- row norms: 16 threads per row, 48 elements each, then 16-way reduce ----
  {
    int r = tid & 15, seg = tid >> 4;
    float ss = 0.f;
    const float* row = &ldsx[r * LROW + seg * 48];
    for (int j = 0; j < 48; ++j) { float v = row[j]; ss = fmaf(v, v, ss); }
    part[r * 16 + seg] = ss;
  }
  __syncthreads();
  if (tid < ROWS) {
    float ss = 0.f;
    for (int s = 0; s < 16; ++s) ss += part[tid * 16 + s];
    float n = fmaxf(sqrtf(ss), 1e-8f);
    snorm[tid] = n;
    sinv[tid]  = 1.0f / n;
    normo[rowBase + tid] = n;
  }
  __syncthreads();

  // ---- prescale tile to x_unit (fold 1/norm out of the hot loop) ----
  {
    float4* l4 = (float4*)ldsx;
    for (int k = tid; k < ROWS * (DD / 4); k += 256) {
      int r = k / (DD / 4), c4 = k % (DD / 4);
      float s = sinv[r];
      float4 v = l4[r * (LROW / 4) + c4];
      v.x *= s; v.y *= s; v.z *= s; v.w *= s;
      l4[r * (LROW / 4) + c4] = v;
    }
  }
  __syncthreads();

  // ---- per-wave group loop: 8 waves x 32 groups, 2 WMMAs per group, branchless ----
  const int  wave = tid >> 5;
  const int  lane = tid & 31;
  const int  col  = lane & 15;        // x-row within tile / B column
  const bool hi   = lane >= 16;       // hi half-wave supplies K=2,3 of A/B operands
  const int  mrow = col;              // A row index; rows >=3 clamp onto the zero entry
  const float nrm = snorm[col];
  // per-lane, loop-invariant offsets into the packed A table (floats)
  const int laneOff = (hi ? 8 : 0) + ((mrow < 3) ? mrow : 3) * 2;
  const float* __restrict__ bxp = &ldsx[col * LROW + (hi ? 2 : 0)];
  const float* __restrict__ byp = &ldsx[col * LROW + 1];
  float* __restrict__ xo = xhat + (rowBase + col) * (long long)DD;
  float* __restrict__ io = idxo + (rowBase + col) * (long long)DD;

#pragma unroll 2
  for (int gi = 0; gi < GG / 8; ++gi) {
    const int g    = wave * (GG / 8) + gi;
    const int gb   = g * 32;
    const int base = 3 * g;

    v2f aF = *(const v2f*)(ldsP + gb + laneOff);         // ds_load_b64
    v2f aB = *(const v2f*)(ldsP + gb + 16 + laneOff);    // ds_load_b64

    v2f b;
    b.x = bxp[base];                    // lo: x_u[col][3g+0] ; hi: x_u[col][3g+2]
    float byv = byp[base];              // x_u[col][3g+1]
    b.y = hi ? 0.f : byv;

    v8f czero = {0.f, 0.f, 0.f, 0.f, 0.f, 0.f, 0.f, 0.f};
    v8f d = __builtin_amdgcn_wmma_f32_16x16x4_f32(
        false, aF, false, b, (short)0, czero, false, false);

    // D[m][r]: VGPR m (0..2), lane r (0..15) = rotated comp m of x-row r
    float q0, q1, q2, i0, i1, i2;
    quantize1(d[0], c0, cstep, q0, i0);
    quantize1(d[1], c0, cstep, q1, i1);
    quantize1(d[2], c0, cstep, q2, i2);

    // backward B: comp2 of row n must move to lane 16+n
    float q2s = __shfl_xor(q2, 16, 32);
    v2f b2;
    b2.x = hi ? q2s : q0;
    b2.y = hi ? 0.f : q1;

    v8f d2 = __builtin_amdgcn_wmma_f32_16x16x4_f32(
        false, aB, false, b2, (short)0, czero, false, false);

    if (!hi) {
      io[base + 0] = i0;  io[base + 1] = i1;  io[base + 2] = i2;
      xo[base + 0] = d2[0] * nrm;
      xo[base + 1] = d2[1] * nrm;
      xo[base + 2] = d2[2] * nrm;
    }
  }
}

extern "C" void kernel_launch(void* const* d_in, const int* in_sizes, int n_in,
                              void* d_out, int out_size, void* d_ws, size_t ws_size,
                              hipStream_t stream) {
  const float* x         = (const float*)d_in[0];   // [N, 768]
  const float* centroids = (const float*)d_in[1];   // [16]
  const float* rotors    = (const float*)d_in[2];   // [G, 8]

  const int N = in_sizes[0] / DD;       // 8192
  const int G = in_sizes[2] / 8;        // 256

  float* out   = (float*)d_out;
  float* xhat  = out;                                   // [N, 768]
  float* idxo  = out + (size_t)N * DD;                  // [N, 768] (indices as float)
  float* normo = out + (size_t)N * DD * 2;              // [N]

  float* Mpack = (float*)d_ws;                          // [G, 32] packed A operands

  rotprep_kernel<<<(G + 255) / 256, 256, 0, stream>>>(rotors, Mpack, G);
  rq_main_kernel<<<N / ROWS, 256, 0, stream>>>(x, centroids, Mpack, xhat, idxo, normo);
}